// CMDF_block_21861383537441
// MI455X (gfx1250) — compile-verified
//
#include <hip/hip_runtime.h>

// Problem constants (match reference)
#define Bn   8
#define Cn   128
#define Hn   64
#define Wn   64
#define HW   (Hn * Wn)          // 4096
#define C2   (2 * Cn)           // 256
#define KKn  9                  // 3x3 taps
#define MID2 64                 // 2*MID
#define CKK  (Cn * KKn)         // 1152
#define PT   (HW / 16)          // 256 pixel tiles per batch

typedef __attribute__((ext_vector_type(2))) float v2f;
typedef __attribute__((ext_vector_type(8))) float v8f;

// ---------------------------------------------------------------------------
// 1) Depthwise 3x3 conv (SAME, groups=C):  Xs = dwconv(X2, static_w)
// ---------------------------------------------------------------------------
__global__ void dw_kernel(const float* __restrict__ X2,
                          const float* __restrict__ Wd,
                          float* __restrict__ Xs) {
    int idx = blockIdx.x * blockDim.x + threadIdx.x;     // over B*C*HW
    if (idx >= Bn * Cn * HW) return;
    int p  = idx % HW;
    int bc = idx / HW;
    int c  = bc % Cn;
    int h = p / Wn, w = p % Wn;
    const float* x  = X2 + (size_t)bc * HW;
    const float* wt = Wd + c * KKn;
    float acc = 0.f;
#pragma unroll
    for (int di = 0; di < 3; ++di) {
        int hh = h + di - 1;
#pragma unroll
        for (int dj = 0; dj < 3; ++dj) {
            int ww = w + dj - 1;
            if ((unsigned)hh < Hn && (unsigned)ww < Wn)
                acc += wt[di * 3 + dj] * x[hh * Wn + ww];
        }
    }
    Xs[idx] = acc;
}

// ---------------------------------------------------------------------------
// 2) Global spatial sum of Fc = [Xs ; Y2] per (b, channel): pooled[b][ch2]
// ---------------------------------------------------------------------------
__global__ void pool_kernel(const float* __restrict__ Xs,
                            const float* __restrict__ Y2,
                            float* __restrict__ pooled) {
    int blk = blockIdx.x;                 // B * 2C blocks
    int b = blk / C2, ch = blk % C2;
    const float* src = (ch < Cn) ? (Xs + ((size_t)b * Cn + ch) * HW)
                                 : (Y2 + ((size_t)b * Cn + (ch - Cn)) * HW);
    __shared__ float red[256];
    float s = 0.f;
    for (int i = threadIdx.x; i < HW; i += 256) s += src[i];
    red[threadIdx.x] = s;
    __syncthreads();
    for (int off = 128; off > 0; off >>= 1) {
        if (threadIdx.x < (unsigned)off) red[threadIdx.x] += red[threadIdx.x + off];
        __syncthreads();
    }
    if (threadIdx.x == 0) pooled[b * C2 + ch] = red[0];
}

// ---------------------------------------------------------------------------
// 3) Context MLP (tiny): ctx1 = w1@mean(Fc); ctx2 = relu(w2@ctx1);
//    chf = w3@ctx2  -> (B, C*9).   Single 256-thread block.
// ---------------------------------------------------------------------------
__global__ void ctx_kernel(const float* __restrict__ pooled,
                           const float* __restrict__ w1,
                           const float* __restrict__ w2,
                           const float* __restrict__ w3,
                           float* __restrict__ chf) {
    __shared__ float ctx1[Bn * MID2];
    __shared__ float ctx2[Bn * MID2];
    int t = threadIdx.x;
    const float inv = 1.0f / (float)HW;
    for (int i = t; i < Bn * MID2; i += 256) {
        int b = i / MID2, o = i % MID2;
        const float* wr = w1 + o * C2;
        const float* pr = pooled + b * C2;
        float acc = 0.f;
        for (int k = 0; k < C2; ++k) acc += wr[k] * pr[k];
        ctx1[i] = acc * inv;
    }
    __syncthreads();
    for (int i = t; i < Bn * MID2; i += 256) {
        int b = i / MID2, o = i % MID2;
        const float* wr = w2 + o * MID2;
        const float* cr = ctx1 + b * MID2;
        float acc = 0.f;
        for (int k = 0; k < MID2; ++k) acc += wr[k] * cr[k];
        ctx2[i] = fmaxf(acc, 0.f);
    }
    __syncthreads();
    for (int i = t; i < Bn * CKK; i += 256) {
        int b = i / CKK, o = i % CKK;
        const float* wr = w3 + o * MID2;
        const float* cr = ctx2 + b * MID2;
        float acc = 0.f;
        for (int k = 0; k < MID2; ++k) acc += wr[k] * cr[k];
        chf[i] = acc;
    }
}

// ---------------------------------------------------------------------------
// 4) Spatial filter GEMM via WMMA fp32:
//    S[b][m][p] = sum_k ws[m][k] * Fc[b][k][p],  m in 0..8 (zero-padded to 16)
//    A (ws, zero-padded) is staged once per block in LDS with row stride 260
//    (-> A-fragment ds_load_b64 hits all 64 banks, conflict-free).
//    K loop is split at the Xs|Y2 boundary: no selects/branches inside.
//    Software-pipelined: fragments for step k+4 are loaded before the WMMA of
//    step k issues, so the wait before WMMA is loadcnt<=2 instead of 0.
//    WMMA layouts (ISA 7.12.2):
//      A 16x4 : lane%16 = M row, (x,y) = K pair, lane/16 selects K{0,1}/{2,3}
//      B 4x16 : lane%16 = N col, (x,y) = K rows, lane/16 selects K half
//      C 16x16: 8 VGPRs, lane/16 adds 8 to M
// ---------------------------------------------------------------------------
#define APAD 260
__global__ void spatial_wmma_kernel(const float* __restrict__ Xs,
                                    const float* __restrict__ Y2,
                                    const float* __restrict__ ws9,
                                    float* __restrict__ S) {
    __shared__ float Apad[16][APAD];
    for (int i = threadIdx.x; i < 16 * C2; i += 256) {
        int r = i >> 8, c = i & 255;
        Apad[r][c] = (r < KKn) ? ws9[r * C2 + c] : 0.f;
    }
    __syncthreads();

    int wave = (blockIdx.x * blockDim.x + threadIdx.x) >> 5;    // 8 waves/block
    int lane = threadIdx.x & 31;
    int b    = wave / PT;
    int p0   = (wave % PT) * 16;
    int lm   = lane & 15;
    int half = lane >> 4;
    int ks   = 2 * half;                                        // lane K skew

    const float* fx = Xs + (size_t)b * Cn * HW + p0 + lm;
    const float* fy = Y2 + (size_t)b * Cn * HW + p0 + lm;

    v8f acc = {};
    // ---- K half 1: rows 0..127 come from Xs ----
    v2f a, bb, an, bn;
    a  = *(const v2f*)&Apad[lm][ks];
    bb.x = fx[(size_t)(ks) * HW];
    bb.y = fx[(size_t)(ks + 1) * HW];
    for (int k = 4; k < Cn; k += 4) {
        int ka = k + ks;
        an   = *(const v2f*)&Apad[lm][ka];
        bn.x = fx[(size_t)ka * HW];
        bn.y = fx[(size_t)(ka + 1) * HW];
        acc = __builtin_amdgcn_wmma_f32_16x16x4_f32(false, a, false, bb,
                                                    (short)0, acc, false, false);
        a = an; bb = bn;
    }
    // ---- K half 2: rows 128..255 come from Y2 ----
    an   = *(const v2f*)&Apad[lm][Cn + ks];
    bn.x = fy[(size_t)(ks) * HW];
    bn.y = fy[(size_t)(ks + 1) * HW];
    acc = __builtin_amdgcn_wmma_f32_16x16x4_f32(false, a, false, bb,
                                                (short)0, acc, false, false);
    a = an; bb = bn;
    for (int k = 4; k < Cn; k += 4) {
        int ka = k + ks;
        an   = *(const v2f*)&Apad[lm][Cn + ka];
        bn.x = fy[(size_t)ka * HW];
        bn.y = fy[(size_t)(ka + 1) * HW];
        acc = __builtin_amdgcn_wmma_f32_16x16x4_f32(false, a, false, bb,
                                                    (short)0, acc, false, false);
        a = an; bb = bn;
    }
    acc = __builtin_amdgcn_wmma_f32_16x16x4_f32(false, a, false, bb,
                                                (short)0, acc, false, false);

    float* Sb = S + (size_t)b * KKn * HW;
#pragma unroll
    for (int r = 0; r < 8; ++r) {
        int row = r + 8 * half;
        if (row < KKn) Sb[(size_t)row * HW + p0 + lm] = acc[r];
    }
}

// ---------------------------------------------------------------------------
// 5) Apply dynamic filter: dyn[b,c,p] = sum_k patch_k(X2) * (chf[b,c,k] + S[b,k,p])
// ---------------------------------------------------------------------------
__global__ void dyn_kernel(const float* __restrict__ X2,
                           const float* __restrict__ chf,
                           const float* __restrict__ S,
                           float* __restrict__ dyn) {
    int idx = blockIdx.x * blockDim.x + threadIdx.x;
    if (idx >= Bn * Cn * HW) return;
    int p  = idx % HW;
    int bc = idx / HW;
    int c  = bc % Cn;
    int b  = bc / Cn;
    int h = p / Wn, w = p % Wn;
    const float* x  = X2 + (size_t)bc * HW;
    const float* cf = chf + ((size_t)b * Cn + c) * KKn;
    const float* Sb = S + (size_t)b * KKn * HW + p;
    float acc = 0.f;
#pragma unroll
    for (int di = 0; di < 3; ++di) {
        int hh = h + di - 1;
#pragma unroll
        for (int dj = 0; dj < 3; ++dj) {
            int ww = w + dj - 1;
            int k  = di * 3 + dj;
            float tap = ((unsigned)hh < Hn && (unsigned)ww < Wn) ? x[hh * Wn + ww] : 0.f;
            acc += tap * (cf[k] + Sb[(size_t)k * HW]);
        }
    }
    dyn[idx] = acc;
}

// ---------------------------------------------------------------------------
// 6) Fusion GEMM via WMMA fp32:
//    out[b][m][p] = sum_k wf[m][k] * Fc2[b][k][p],  Fc2 = [Xs ; dyn]
//    256-thread block = 8 waves sharing one 16-pixel column tile; the full
//    256x16 B tile (16KB) is staged cooperatively into LDS (coalesced float4
//    loads, Xs|dyn select resolved once per thread). Inner loop: aligned
//    global_load_b64 of wf (A) + two conflict-free ds_load_b32 (B) + WMMA,
//    with one-step prefetch pipelining.
// ---------------------------------------------------------------------------
__global__ void fusion_wmma_kernel(const float* __restrict__ Xs,
                                   const float* __restrict__ dyn,
                                   const float* __restrict__ wf,
                                   float* __restrict__ out) {
    __shared__ float Ft[C2][16];                      // B tile: [channel][pixel]
    int blk = blockIdx.x;                             // B*PT blocks
    int b   = blk / PT;
    int p0  = (blk % PT) * 16;
    {   // cooperative stage: thread t owns channel row t
        int ch = threadIdx.x;
        const float* src = (ch < Cn)
            ? (Xs  + ((size_t)b * Cn + ch) * HW + p0)
            : (dyn + ((size_t)b * Cn + (ch - Cn)) * HW + p0);
        const float4* s4 = (const float4*)src;
        float4* d4 = (float4*)&Ft[ch][0];
        d4[0] = s4[0]; d4[1] = s4[1]; d4[2] = s4[2]; d4[3] = s4[3];
    }
    __syncthreads();

    int mt   = threadIdx.x >> 5;                      // wave id = m-tile (0..7)
    int lane = threadIdx.x & 31;
    int lm   = lane & 15;
    int half = lane >> 4;
    int ks   = 2 * half;
    int m0   = mt * 16;
    const float* wr = wf + (size_t)(m0 + lm) * C2;    // 8B-aligned (ks even)

    v8f acc = {};
    v2f a, bb, an, bn;
    a  = *(const v2f*)(wr + ks);
    bb.x = Ft[ks][lm];
    bb.y = Ft[ks + 1][lm];
    for (int k = 4; k < C2; k += 4) {
        int ka = k + ks;
        an   = *(const v2f*)(wr + ka);
        bn.x = Ft[ka][lm];
        bn.y = Ft[ka + 1][lm];
        acc = __builtin_amdgcn_wmma_f32_16x16x4_f32(false, a, false, bb,
                                                    (short)0, acc, false, false);
        a = an; bb = bn;
    }
    acc = __builtin_amdgcn_wmma_f32_16x16x4_f32(false, a, false, bb,
                                                (short)0, acc, false, false);

    float* ob = out + (size_t)b * Cn * HW;
#pragma unroll
    for (int r = 0; r < 8; ++r) {
        int row = m0 + r + 8 * half;
        ob[(size_t)row * HW + p0 + lm] = acc[r];
    }
}

// ---------------------------------------------------------------------------
extern "C" void kernel_launch(void* const* d_in, const int* in_sizes, int n_in,
                              void* d_out, int out_size, void* d_ws, size_t ws_size,
                              hipStream_t stream) {
    const float* X2 = (const float*)d_in[0];
    const float* Y2 = (const float*)d_in[1];
    const float* Wd = (const float*)d_in[2];   // static_w (C,1,3,3)
    const float* w1 = (const float*)d_in[3];   // (64,256)
    const float* w2 = (const float*)d_in[4];   // (64,64)
    const float* w3 = (const float*)d_in[5];   // (1152,64)
    const float* ws9 = (const float*)d_in[6];  // (9,256)
    const float* wf = (const float*)d_in[7];   // (128,256)
    float* out = (float*)d_out;

    // workspace layout (floats)
    float* wsF    = (float*)d_ws;
    float* Xs     = wsF;                                   // B*C*HW
    float* dyn    = Xs  + (size_t)Bn * Cn * HW;            // B*C*HW
    float* S      = dyn + (size_t)Bn * Cn * HW;            // B*9*HW
    float* pooled = S   + (size_t)Bn * KKn * HW;           // B*256
    float* chf    = pooled + (size_t)Bn * C2;              // B*1152

    const int NTOT = Bn * Cn * HW;                         // 4,194,304

    // 1) depthwise conv
    dw_kernel<<<(NTOT + 255) / 256, 256, 0, stream>>>(X2, Wd, Xs);
    // 2) pooled Fc sums
    pool_kernel<<<Bn * C2, 256, 0, stream>>>(Xs, Y2, pooled);
    // 3) context MLP -> channel filter
    ctx_kernel<<<1, 256, 0, stream>>>(pooled, w1, w2, w3, chf);
    // 4) spatial filter GEMM (WMMA): 8 waves/block, B*HW/16 waves total
    spatial_wmma_kernel<<<(Bn * PT) / 8, 256, 0, stream>>>(Xs, Y2, ws9, S);
    // 5) dynamic filter application
    dyn_kernel<<<(NTOT + 255) / 256, 256, 0, stream>>>(X2, chf, S, dyn);
    // 6) fusion GEMM (WMMA)
    fusion_wmma_kernel<<<Bn * PT, 256, 0, stream>>>(Xs, dyn, wf, out);
}